// GraphConv_19439021981794
// MI455X (gfx1250) — compile-verified
//
#include <hip/hip_runtime.h>

#define FEAT 128

typedef __attribute__((ext_vector_type(2))) float v2f;
typedef __attribute__((ext_vector_type(8))) float v8f;

// ---------------------------------------------------------------------------
// Zero the workspace (agg + deg) every launch with b128 stores.
// Caller guarantees the float count is a multiple of 4.
// ---------------------------------------------------------------------------
__global__ void gc_zero_kernel(float4* __restrict__ p, long n4) {
  long i = (long)blockIdx.x * blockDim.x + threadIdx.x;
  long stride = (long)gridDim.x * blockDim.x;
  float4 z = {0.0f, 0.0f, 0.0f, 0.0f};
  for (; i < n4; i += stride) p[i] = z;
}

// ---------------------------------------------------------------------------
// Edge scatter: one wave32 per edge. Lane k moves float4 at feature 4k
// (b128 global load + 4x global_atomic_add_f32). Lane 0 counts the degree.
// __hip_atomic_fetch_add(relaxed, agent) forces the hardware f32 atomic
// (GLOBAL_ATOMIC_ADD_F32) instead of a CAS retry loop — this phase is the
// runtime-dominant one (~655 MB of L2-resident gather+atomic traffic).
// ---------------------------------------------------------------------------
__device__ __forceinline__ void fadd_agent(float* p, float v) {
  __hip_atomic_fetch_add(p, v, __ATOMIC_RELAXED, __HIP_MEMORY_SCOPE_AGENT);
}

__global__ void gc_scatter_kernel(const float* __restrict__ x,
                                  const int* __restrict__ src,
                                  const int* __restrict__ dst,
                                  float* __restrict__ agg,
                                  float* __restrict__ deg,
                                  int E) {
  int tid  = blockIdx.x * blockDim.x + threadIdx.x;
  int e    = tid >> 5;
  int lane = tid & 31;
  if (e >= E) return;
  int s = src[e];
  int d = dst[e];
  const float4 v = ((const float4*)(x + (long)s * FEAT))[lane];
  float* ad = agg + (long)d * FEAT + lane * 4;
  fadd_agent(ad + 0, v.x);
  fadd_agent(ad + 1, v.y);
  fadd_agent(ad + 2, v.z);
  fadd_agent(ad + 3, v.w);
  if (lane == 0) fadd_agent(deg + d, 1.0f);
}

// ---------------------------------------------------------------------------
// Fused (agg/deg) @ W^T + b, ReLU — exact fp32 via V_WMMA_F32_16X16X4_F32.
//   out[m][n] = relu( (1/deg[m]) * sum_k agg[m][k]*W[n][k] + b[n] )
// Degree normalization is distributive over the K-sum, so it is applied to
// the accumulator at writeback: the K-loop is a pure load+WMMA stream.
//
// Block = 256 threads = 8 waves. Block owns rows [16*blockIdx, +16);
// wave w owns output columns [16*w, +16). K=128 -> 32 accumulating WMMAs.
//
// ISA fragment layouts (CDNA5 ISA):
//   A 16x4 f32 : lane l holds A[l&15][k0 + 2*(l>>4) + {0,1}] in 2 VGPRs
//   B  4x16    : lane l holds B[k0 + 2*(l>>4) + {0,1}][l&15]  (= W[n][k])
//   C/D 16x16  : vgpr i, lane l -> row (l<16 ? i : i+8), col l&15
// EXEC is all-ones: M=100000 is an exact multiple of 16, no divergence.
// ---------------------------------------------------------------------------
__global__ void gc_gemm_kernel(const float* __restrict__ agg,
                               const float* __restrict__ deg,
                               const float* __restrict__ W,
                               const float* __restrict__ bias,
                               float* __restrict__ out) {
  const int mblk  = blockIdx.x;          // 16-row block
  const int ntile = threadIdx.x >> 5;    // wave id 0..7 -> 16-col tile
  const int lane  = threadIdx.x & 31;
  const int half  = lane >> 4;           // 0: K+{0,1}, 1: K+{2,3}
  const int l15   = lane & 15;

  const int arow_i = mblk * 16 + l15;    // A row this lane reads
  const int gcol   = ntile * 16 + l15;   // W row / output column

  const float* __restrict__ arow = agg + (long)arow_i * FEAT;
  const float* __restrict__ wrow = W   + (long)gcol   * FEAT;

  v8f c = {};
#pragma unroll
  for (int k0 = 0; k0 < FEAT; k0 += 4) {
    const int kk = k0 + half * 2;
    v2f a, b;
    a.x = arow[kk];
    a.y = arow[kk + 1];
    b.x = wrow[kk];
    b.y = wrow[kk + 1];
    // 8 args: (neg_a, A, neg_b, B, c_mod, C, reuse_a, reuse_b)
    c = __builtin_amdgcn_wmma_f32_16x16x4_f32(
        false, a, false, b, (short)0, c, false, false);
  }

  const float bb = bias[gcol];
#pragma unroll
  for (int i = 0; i < 8; ++i) {
    const int m = mblk * 16 + i + 8 * half;        // C/D row for this vgpr+lane
    const float rdeg = 1.0f / fmaxf(deg[m], 1.0f); // lane-uniform per 16-half
    const float v = fmaf(c[i], rdeg, bb);
    out[(long)m * FEAT + gcol] = fmaxf(v, 0.0f);
  }
}

// ---------------------------------------------------------------------------
// Launch: zero ws -> scatter -> wmma gemm, all on `stream` (graph-safe).
// Inputs: x [N,128] f32, edge_index [2,E] int, W [128,128] f32, b [128] f32.
// ---------------------------------------------------------------------------
extern "C" void kernel_launch(void* const* d_in, const int* in_sizes, int n_in,
                              void* d_out, int out_size, void* d_ws, size_t ws_size,
                              hipStream_t stream) {
  const float* x  = (const float*)d_in[0];
  const int*   ei = (const int*)d_in[1];
  const float* W  = (const float*)d_in[2];
  const float* b  = (const float*)d_in[3];
  float*       out = (float*)d_out;

  const int N = in_sizes[0] / FEAT;   // 100000
  const int E = in_sizes[1] / 2;      // 640000

  float* agg = (float*)d_ws;                  // N*128 floats (51.2 MB)
  float* deg = agg + (size_t)N * FEAT;        // N floats

  const long ztotal4 = ((long)N * FEAT + N) / 4;  // N*(FEAT+1) divisible by 4
  gc_zero_kernel<<<2048, 256, 0, stream>>>((float4*)agg, ztotal4);

  const long sthreads = (long)E * 32;         // one wave per edge
  gc_scatter_kernel<<<(int)((sthreads + 255) / 256), 256, 0, stream>>>(
      x, ei, ei + E, agg, deg, E);

  gc_gemm_kernel<<<N / 16, 256, 0, stream>>>(agg, deg, W, b, out);
}